// LSTM_57784490000969
// MI455X (gfx1250) — compile-verified
//
#include <hip/hip_runtime.h>
#include <stdint.h>

#define TLEN 600
#define TS 16          // timesteps per TDM tile (power of 2 so TDM pad fires at row end)
#define ROWSTRIDE 17   // 16 data dwords + 1 pad dword  -> conflict-free LDS reads
#define WAVES_PER_BLOCK 4
#define THREADS (WAVES_PER_BLOCK * 32)

typedef unsigned int u32;
typedef __attribute__((ext_vector_type(4))) u32 u32x4;
typedef __attribute__((ext_vector_type(8))) int i32x8;
typedef __attribute__((ext_vector_type(4))) int i32x4;
typedef __attribute__((ext_vector_type(2))) float v2f;
typedef __attribute__((ext_vector_type(8))) float v8f;

__device__ __forceinline__ float fexp2(float x) { return __builtin_amdgcn_exp2f(x); }
__device__ __forceinline__ float frcp(float x)  { return __builtin_amdgcn_rcpf(x); }
// sigmoid(z) = 1 / (1 + 2^(-z*log2e))
__device__ __forceinline__ float sigm(float z) {
  return frcp(1.0f + fexp2(-1.4426950408889634f * z));
}
// tanh(z) = 1 - 2 / (1 + 2^(2z*log2e))
__device__ __forceinline__ float tanh_fast(float z) {
  return 1.0f - 2.0f * frcp(1.0f + fexp2(2.8853900817779268f * z));
}

// Tensor Data Mover: pull a [32 samples x TS timesteps] tile of x into LDS,
// padded by hardware to ROWSTRIDE dwords per row. 2D descriptor:
//   dim0 (contiguous) = time, stride 600; dim1 = 32 sample rows.
// rem_cols = remaining valid timesteps (OOB columns read as zero).
__device__ __forceinline__ void tdm_load_tile(const float* gptr, u32 lds_off, int rem_cols) {
  unsigned long long ga = (unsigned long long)(uintptr_t)gptr;
  u32x4 g0;
  g0.x = 1u;                                            // count=1 (valid descriptor)
  g0.y = lds_off;                                       // lds_addr (bytes)
  g0.z = (u32)ga;                                       // global_addr[31:0]
  g0.w = (u32)((ga >> 32) & 0x1FFFFFFull) | (2u << 30); // global_addr[56:32] | type=2
  i32x8 g1;
  g1[0] = (2 << 16)     // data_size = 4B
        | (1 << 20)     // pad_enable
        | (3 << 22)     // pad_interval code 3 = every 16 dwords
        | (0 << 25);    // pad_amount  code 0 = 1 dword
  g1[1] = (rem_cols & 0xFFFF) << 16;                    // tensor_dim0[15:0]
  g1[2] = ((rem_cols >> 16) & 0xFFFF) | (32 << 16);     // tensor_dim0[31:16] | tensor_dim1=32
  g1[3] = (TS << 16);                                   // tensor_dim1[31:16]=0 | tile_dim0=TS
  g1[4] = 32;                                           // tile_dim1=32, tile_dim2=0 (2D)
  g1[5] = TLEN;                                         // tensor_dim0_stride[31:0] = 600
  g1[6] = 0;                                            // stride[47:32]=0, dim1_stride lo=0
  g1[7] = 0;
  i32x4 z4 = {0, 0, 0, 0};                              // groups 2/3 unused (2D tile)
  i32x8 z8 = {0, 0, 0, 0, 0, 0, 0, 0};                  // extra group (clang-23 form)
  __builtin_amdgcn_tensor_load_to_lds(g0, g1, z4, z4, z8, 0);
}

__global__ __launch_bounds__(THREADS, 1) void lstm_fused(
    const float* __restrict__ x,     const float* __restrict__ w_ih,
    const float* __restrict__ w_hh,  const float* __restrict__ b_ih,
    const float* __restrict__ b_hh,  const float* __restrict__ fc_w,
    const float* __restrict__ fc_b,  const float* __restrict__ out_w,
    const float* __restrict__ out_b, float* __restrict__ out)
{
  __shared__ float xtile[WAVES_PER_BLOCK][2][32 * ROWSTRIDE];

  const int lane  = threadIdx.x & 31;
  const int wave  = threadIdx.x >> 5;
  const int sbase = (blockIdx.x * WAVES_PER_BLOCK + wave) * 32;

  // All weights live in registers (wave-uniform values; ~145 scalars).
  float Wih[20], Bb[20], Whh[20][5];
#pragma unroll
  for (int g = 0; g < 20; ++g) {
    Wih[g] = w_ih[g];
    Bb[g]  = b_ih[g] + b_hh[g];
#pragma unroll
    for (int k = 0; k < 5; ++k) Whh[g][k] = w_hh[g * 5 + k];
  }

  float h[5] = {0, 0, 0, 0, 0};
  float c[5] = {0, 0, 0, 0, 0};

  const float* xrow0 = x + (size_t)sbase * TLEN;
  u32 lds_base[2];
  lds_base[0] = (u32)(uintptr_t)&xtile[wave][0][0];
  lds_base[1] = (u32)(uintptr_t)&xtile[wave][1][0];

  // Prime buffer 0, then double-buffer: issue next tile, wait current (in-order).
  tdm_load_tile(xrow0, lds_base[0], TLEN);
  int buf = 0;
  for (int t0 = 0; t0 < TLEN; t0 += TS) {
    const int nxt = t0 + TS;
    if (nxt < TLEN) {
      tdm_load_tile(xrow0 + nxt, lds_base[buf ^ 1], TLEN - nxt);
      __builtin_amdgcn_s_wait_tensorcnt(1);   // current tile done, next in flight
    } else {
      __builtin_amdgcn_s_wait_tensorcnt(0);
    }
    const float* myrow = &xtile[wave][buf][lane * ROWSTRIDE];
    const int lim = (TLEN - t0 < TS) ? (TLEN - t0) : TS;
    for (int tt = 0; tt < lim; ++tt) {
      const float xv = myrow[tt];             // ds_load, conflict-free (stride 17)
      float gate[20];
#pragma unroll
      for (int g = 0; g < 20; ++g) {
        float a = fmaf(xv, Wih[g], Bb[g]);
#pragma unroll
        for (int k = 0; k < 5; ++k) a = fmaf(h[k], Whh[g][k], a);
        gate[g] = a;
      }
#pragma unroll
      for (int k = 0; k < 5; ++k) {
        const float ig = sigm(gate[k]);
        const float fg = sigm(gate[5 + k]);
        const float gg = tanh_fast(gate[10 + k]);
        const float og = sigm(gate[15 + k]);
        c[k] = fmaf(fg, c[k], ig * gg);
        h[k] = og * tanh_fast(c[k]);
      }
    }
    buf ^= 1;
  }

  // ---- Head: out = sigmoid(h . (out_w@fc_w) + (out_w@fc_b + out_b)) ----
  float v[5] = {0, 0, 0, 0, 0};
  float bp = out_b[0];
#pragma unroll
  for (int j = 0; j < 15; ++j) {
    const float ow = out_w[j];
    bp = fmaf(ow, fc_b[j], bp);
#pragma unroll
    for (int k = 0; k < 5; ++k) v[k] = fmaf(ow, fc_w[j * 5 + k], v[k]);
  }

  // v_wmma_f32_16x16x4_f32: M=16 samples, N=16 (replicated), K chunks
  // [h0..h3] then [h4, 1(bias), 0, 0]. A layout: lanes 0-15 hold K0/K1 of
  // sample m=lane, lanes 16-31 hold K2/K3 of sample m=lane-16.
  const bool lo = lane < 16;
  const float h0x = __shfl_xor(h[0], 16, 32);
  const float h1x = __shfl_xor(h[1], 16, 32);
  const float h2x = __shfl_xor(h[2], 16, 32);
  const float h3x = __shfl_xor(h[3], 16, 32);
  const float h4x = __shfl_xor(h[4], 16, 32);

  v2f b1;  b1.x = lo ? v[0] : v[2];  b1.y = lo ? v[1] : v[3];
  v2f b2;  b2.x = lo ? v[4] : 0.0f;  b2.y = lo ? bp   : 0.0f;

  v2f aA;  aA.x  = lo ? h[0] : h2x;  aA.y  = lo ? h[1] : h3x;  // samples 0..15
  v2f aA2; aA2.x = lo ? h[4] : 0.0f; aA2.y = lo ? 1.0f : 0.0f;
  v2f aB;  aB.x  = lo ? h0x : h[2];  aB.y  = lo ? h1x : h[3];  // samples 16..31
  v2f aB2; aB2.x = lo ? h4x : 0.0f;  aB2.y = lo ? 1.0f : 0.0f;

  v8f acc0 = {0, 0, 0, 0, 0, 0, 0, 0};
  acc0 = __builtin_amdgcn_wmma_f32_16x16x4_f32(false, aA,  false, b1, (short)0, acc0, false, false);
  acc0 = __builtin_amdgcn_wmma_f32_16x16x4_f32(false, aA2, false, b2, (short)0, acc0, false, false);
  v8f acc1 = {0, 0, 0, 0, 0, 0, 0, 0};
  acc1 = __builtin_amdgcn_wmma_f32_16x16x4_f32(false, aB,  false, b1, (short)0, acc1, false, false);
  acc1 = __builtin_amdgcn_wmma_f32_16x16x4_f32(false, aB2, false, b2, (short)0, acc1, false, false);

  // D layout: VGPR r holds sample m=r (lanes 0-15) / m=r+8 (lanes 16-31);
  // all 16 columns identical, so lanes 0 and 16 write the results.
  if (lane == 0) {
#pragma unroll
    for (int r = 0; r < 8; ++r) {
      out[sbase + r]      = sigm(acc0[r]);
      out[sbase + 16 + r] = sigm(acc1[r]);
    }
  }
  if (lane == 16) {
#pragma unroll
    for (int r = 0; r < 8; ++r) {
      out[sbase + 8 + r]  = sigm(acc0[r]);
      out[sbase + 24 + r] = sigm(acc1[r]);
    }
  }
}

extern "C" void kernel_launch(void* const* d_in, const int* in_sizes, int n_in,
                              void* d_out, int out_size, void* d_ws, size_t ws_size,
                              hipStream_t stream) {
  const float* x     = (const float*)d_in[0];
  const float* w_ih  = (const float*)d_in[1];
  const float* w_hh  = (const float*)d_in[2];
  const float* b_ih  = (const float*)d_in[3];
  const float* b_hh  = (const float*)d_in[4];
  const float* fc_w  = (const float*)d_in[5];
  const float* fc_b  = (const float*)d_in[6];
  const float* out_w = (const float*)d_in[7];
  const float* out_b = (const float*)d_in[8];
  float* out = (float*)d_out;

  const int N = in_sizes[0] / TLEN;                  // 32768
  const int blocks = N / (WAVES_PER_BLOCK * 32);     // 256 blocks of 4 waves
  lstm_fused<<<blocks, THREADS, 0, stream>>>(x, w_ih, w_hh, b_ih, b_hh,
                                             fc_w, fc_b, out_w, out_b, out);
}